// ETKFWeightsModule_27324581937270
// MI455X (gfx1250) — compile-verified
//
#include <hip/hip_runtime.h>

// ---------------------------------------------------------------------------
// ETKF weights: one workgroup (128 threads, 4 waves) per batch element.
//  P load               : global_load_async_to_lds_b128 (ASYNCcnt path)
//  S = P P^T            : WMMA f32 16x16x4 (exact fp32), 3x3 tiles, K=64
//  eigh(S)              : parallel cyclic Jacobi, closed-form tournament
//                         pairing, wave-owned pairs -> 2 barriers/round
//  cov / w_perts        : U diag(d) U^T via two WMMA GEMMs, K=48 (zero-padded)
//  w_mean / weights     : small dots + broadcast add
// ---------------------------------------------------------------------------

typedef __attribute__((ext_vector_type(2))) float v2f;
typedef __attribute__((ext_vector_type(8))) float v8f;

#define N_ENS     40
#define N_OBS     64
#define NP        48     // padded matrix dim (3 x 16)
#define MS        49     // LDS row stride for 48x48 matrices (odd -> conflict free)
#define SP_STRIDE 68     // LDS row stride for P (16B aligned rows, 4r mod 64 distinct)
#define NSWEEPS   8

// Round-robin (circle method) pairing for 40 indices, 39 rounds x 20 pairs.
__device__ __forceinline__ void pair_rk(int r, int k, int& p, int& q) {
  int a, b;
  if (k == 0) { a = 39; b = r; }
  else        { a = (r + k) % 39; b = (r + 39 - k) % 39; }
  p = a < b ? a : b;
  q = a < b ? b : a;
}

// C[48x48] = A[48x48] * B^T  (B row-major, read as B[n][k]), K = 48.
__device__ __forceinline__ void wmma_gemm_ABt(const float* A, const float* Bm,
                                              float* C, int lane, int wave) {
  for (int tile = wave; tile < 9; tile += 4) {          // wave-uniform tiles
    const int ti = (tile / 3) * 16, tj = (tile % 3) * 16;
    const int m  = ti + (lane & 15);
    const int n  = tj + (lane & 15);
    const int kh = (lane >> 4) * 2;
    v8f acc = {};
    for (int k0 = 0; k0 < NP; k0 += 4) {
      v2f av, bv;
      av.x = A [m * MS + k0 + kh];
      av.y = A [m * MS + k0 + kh + 1];
      bv.x = Bm[n * MS + k0 + kh];
      bv.y = Bm[n * MS + k0 + kh + 1];
      acc = __builtin_amdgcn_wmma_f32_16x16x4_f32(false, av, false, bv,
                                                  (short)0, acc, false, false);
    }
#pragma unroll
    for (int r = 0; r < 8; ++r)
      C[(ti + r + ((lane >> 4) << 3)) * MS + tj + (lane & 15)] = acc[r];
  }
}

// S[48x48] = P[48x64] * P^T, K = 64 (rows 40..47 of P are zero-filled).
__device__ __forceinline__ void wmma_syrk_P(const float* P, float* S,
                                            int lane, int wave) {
  for (int tile = wave; tile < 9; tile += 4) {
    const int ti = (tile / 3) * 16, tj = (tile % 3) * 16;
    const int m  = ti + (lane & 15);
    const int n  = tj + (lane & 15);
    const int kh = (lane >> 4) * 2;
    v8f acc = {};
    for (int k0 = 0; k0 < N_OBS; k0 += 4) {
      v2f av, bv;
      av.x = P[m * SP_STRIDE + k0 + kh];
      av.y = P[m * SP_STRIDE + k0 + kh + 1];
      bv.x = P[n * SP_STRIDE + k0 + kh];
      bv.y = P[n * SP_STRIDE + k0 + kh + 1];
      acc = __builtin_amdgcn_wmma_f32_16x16x4_f32(false, av, false, bv,
                                                  (short)0, acc, false, false);
    }
#pragma unroll
    for (int r = 0; r < 8; ++r)
      S[(ti + r + ((lane >> 4) << 3)) * MS + tj + (lane & 15)] = acc[r];
  }
}

__global__ __launch_bounds__(128)
void etkf_weights_kernel(const float* __restrict__ perts,
                         const float* __restrict__ obs,
                         const float* __restrict__ inf,
                         float* __restrict__ out_weights,
                         float* __restrict__ out_wmean,
                         float* __restrict__ out_wperts,
                         float* __restrict__ out_cov) {
  __shared__ float sP[NP * SP_STRIDE];   // P padded to 48 rows
  __shared__ float sA[NP * MS];          // Gram -> eigen workspace -> w_perts
  __shared__ float sV[NP * MS];          // eigenvectors
  __shared__ float sW[NP * MS];          // V * diag(.) scratch
  __shared__ float sC[NP * MS];          // cov_analysed
  __shared__ float sObs[N_OBS];
  __shared__ float sKobs[N_ENS];
  __shared__ float sWmean[N_ENS];
  __shared__ float sRc[20], sRs[20];     // per-pair rotation c,s
  __shared__ float sEinv[NP], sSq[NP];

  const int tid  = threadIdx.x;
  const int lane = tid & 31;
  const int wave = tid >> 5;
  const int b    = blockIdx.x;

  // ---- async-stage P (40x64 fp32) into LDS, 16B per lane per issue ---------
  const float4* gP = (const float4*)(perts + (size_t)b * N_ENS * N_OBS);
  for (int i = tid; i < (N_ENS * N_OBS) / 4; i += 128) {   // 5 uniform iters
    const int row = i >> 4, c4 = i & 15;                   // 16 float4 per row
    const unsigned ldsoff = (unsigned)(size_t)&sP[row * SP_STRIDE + c4 * 4];
    asm volatile("global_load_async_to_lds_b128 %0, %1, off"
                 :: "v"(ldsoff), "v"(gP + i) : "memory");
  }
  // zero-pad rows 40..47 (plain DS stores, disjoint region)
  for (int i = tid; i < 8 * 16; i += 128) {
    const int row = N_ENS + (i >> 4), c4 = i & 15;
    *(float4*)&sP[row * SP_STRIDE + c4 * 4] = make_float4(0.f, 0.f, 0.f, 0.f);
  }
  if (tid < N_OBS) sObs[tid] = obs[(size_t)b * N_OBS + tid];
  for (int i = tid; i < NP * NP; i += 128) {               // V = I (padded 0)
    const int r = i / NP, c = i % NP;
    sV[r * MS + c] = (r == c && r < N_ENS) ? 1.f : 0.f;
  }
  asm volatile("s_wait_asynccnt 0x0" ::: "memory");        // async P complete
  __syncthreads();

  // ---- kernel_obs = P * obs^T  (40 dot-64s) --------------------------------
  if (tid < N_ENS) {
    float acc = 0.f;
    for (int j = 0; j < N_OBS; ++j) acc += sP[tid * SP_STRIDE + j] * sObs[j];
    sKobs[tid] = acc;
  }

  // ---- S = P P^T  via WMMA --------------------------------------------------
  wmma_syrk_P(sP, sA, lane, wave);
  __syncthreads();

  // ---- cyclic Jacobi eigendecomposition of sA (40x40), V accumulated -------
  // Wave w owns pairs {w, w+4, ..., w+16}. Rows of distinct pairs are
  // disjoint; cols of distinct pairs are disjoint. Rotation params are only
  // consumed by the owning wave (LDS ops are in-order within a wave), so only
  // row->col and col->next-round transitions need workgroup barriers.
  for (int sweep = 0; sweep < NSWEEPS; ++sweep) {
    for (int round = 0; round < 39; ++round) {
      if (lane < 5) {                       // this wave's 5 rotations
        const int k = wave + 4 * lane;
        int p, q; pair_rk(round, k, p, q);
        const float app = sA[p * MS + p];
        const float aqq = sA[q * MS + q];
        const float apq = sA[p * MS + q];
        float c = 1.f, s = 0.f;
        if (fabsf(apq) > 1e-12f) {
          const float theta = (aqq - app) / (2.f * apq);
          const float t = copysignf(1.f, theta) /
                          (fabsf(theta) + sqrtf(1.f + theta * theta));
          c = rsqrtf(1.f + t * t);
          s = t * c;
        }
        sRc[k] = c; sRs[k] = s;
      }
      __builtin_amdgcn_wave_barrier();      // in-wave ordering (DS is in-order)
      // row update: own 5 pairs, rows p,q over cols 0..39 (pair rows disjoint)
      for (int t = lane; t < 5 * N_ENS; t += 32) {
        const int k = wave + 4 * (t / N_ENS);
        const int j = t % N_ENS;
        int p, q; pair_rk(round, k, p, q);
        const float c = sRc[k], s = sRs[k];
        const float ap = sA[p * MS + j], aq = sA[q * MS + j];
        sA[p * MS + j] = c * ap - s * aq;
        sA[q * MS + j] = s * ap + c * aq;
      }
      __syncthreads();
      // column update on A, plus V = V*J (pair columns disjoint across waves)
      for (int t = lane; t < 2 * 5 * N_ENS; t += 32) {
        const int grp = t / N_ENS;          // 0..9
        const int k = wave + 4 * (grp % 5);
        const int which = grp / 5;
        const int j = t % N_ENS;
        int p, q; pair_rk(round, k, p, q);
        const float c = sRc[k], s = sRs[k];
        float* M = which ? sV : sA;
        const float mp = M[j * MS + p], mq = M[j * MS + q];
        M[j * MS + p] = c * mp - s * mq;
        M[j * MS + q] = s * mp + c * mq;
      }
      __syncthreads();
    }
  }

  // ---- eigenvalue post-processing ------------------------------------------
  const float reg = (float)(N_ENS - 1) / inf[0];
  if (tid < NP) {
    float e = (tid < N_ENS) ? sA[tid * MS + tid] : 0.f;
    e = fmaxf(e, 0.f) + reg;
    const float einv = 1.f / e;
    sEinv[tid] = (tid < N_ENS) ? einv : 0.f;
    sSq[tid]   = (tid < N_ENS) ? sqrtf((float)(N_ENS - 1) * einv) : 0.f;
  }
  __syncthreads();

  // ---- cov = (V diag(einv)) V^T  via WMMA ----------------------------------
  for (int i = tid; i < NP * NP; i += 128) {
    const int r = i / NP, c = i % NP;
    sW[r * MS + c] = sV[r * MS + c] * sEinv[c];
  }
  __syncthreads();
  wmma_gemm_ABt(sW, sV, sC, lane, wave);
  __syncthreads();

  // ---- w_perts = (V diag(sqrt(39*einv))) V^T  via WMMA (into sA) -----------
  for (int i = tid; i < NP * NP; i += 128) {
    const int r = i / NP, c = i % NP;
    sW[r * MS + c] = sV[r * MS + c] * sSq[c];
  }
  __syncthreads();
  wmma_gemm_ABt(sW, sV, sA, lane, wave);
  __syncthreads();

  // ---- w_mean = cov * kernel_obs -------------------------------------------
  if (tid < N_ENS) {
    float acc = 0.f;
    for (int j = 0; j < N_ENS; ++j) acc += sC[tid * MS + j] * sKobs[j];
    sWmean[tid] = acc;
  }
  __syncthreads();

  // ---- stores: weights | w_mean | w_perts | cov ----------------------------
  const size_t base = (size_t)b * (N_ENS * N_ENS);
  float* gW  = out_weights + base;
  float* gWp = out_wperts  + base;
  float* gC  = out_cov     + base;
  for (int i = tid; i < N_ENS * N_ENS; i += 128) {
    const int m = i / N_ENS, n = i % N_ENS;
    const float wp = sA[m * MS + n];
    gWp[i] = wp;
    gW[i]  = wp + sWmean[m];
    gC[i]  = sC[m * MS + n];
  }
  if (tid < N_ENS) out_wmean[(size_t)b * N_ENS + tid] = sWmean[tid];
}

extern "C" void kernel_launch(void* const* d_in, const int* in_sizes, int n_in,
                              void* d_out, int out_size, void* d_ws, size_t ws_size,
                              hipStream_t stream) {
  const float* perts = (const float*)d_in[0];   // [B,40,64]
  const float* obs   = (const float*)d_in[1];   // [B,1,64]
  const float* inf   = (const float*)d_in[2];   // scalar

  const int B = in_sizes[0] / (N_ENS * N_OBS);  // 32768

  float* out = (float*)d_out;
  float* out_weights = out;                                    // B*1600
  float* out_wmean   = out + (size_t)B * N_ENS * N_ENS;        // B*40
  float* out_wperts  = out_wmean + (size_t)B * N_ENS;          // B*1600
  float* out_cov     = out_wperts + (size_t)B * N_ENS * N_ENS; // B*1600

  etkf_weights_kernel<<<B, 128, 0, stream>>>(perts, obs, inf,
                                             out_weights, out_wmean,
                                             out_wperts, out_cov);
}